// TLoRANetwork_53017076302218
// MI455X (gfx1250) — compile-verified
//
#include <hip/hip_runtime.h>
#include <stdint.h>

// ---------------------------------------------------------------------------
// TLoRA forward for MI455X (gfx1250, wave32).
//   out = x @ W_org^T + (x q^T * lam) p^T - (x bq^T * blam) bp^T
// Folded once (4.3 GFLOP):  W_eff = W_org + p diag(lam) q - bp diag(blam) bq
// Then one fp32 GEMM (550 GFLOP): out[16384x4096] = x @ W_eff^T using
// V_WMMA_F32_16X16X4_F32 (exact fp32 matrix path on CDNA5).
// W_eff (64MB, in d_ws) is L2-resident (192MB L2) -> kernel is WMMA-bound;
// HBM floor ~0.6GB / 23.3TB/s ~= 25us.
// Global->LDS uses CDNA5 async-to-LDS (ASYNCcnt) with LDS double buffering:
// next slab streams in while WMMAs run on the current slab; 1 barrier/iter.
// ---------------------------------------------------------------------------

typedef __attribute__((ext_vector_type(2))) float v2f;
typedef __attribute__((ext_vector_type(8))) float v8f;

#define IN_DIM  4096
#define OUT_DIM 4096
#define RANK    64

#define MT 128           // block tile M
#define NT 128           // block tile N
#define KT 32            // K slab staged in LDS
#define LDS_STRIDE 36    // floats per LDS row: 144B (16B aligned), bank-conflict-free

__device__ __forceinline__ void async_g2l_b128(const void* g, void* l) {
  // Generic 'shared' pointer: low 32 bits are the LDS byte offset.
  unsigned lofs = (unsigned)(uint64_t)l;
  asm volatile("global_load_async_to_lds_b128 %0, %1, off"
               :: "v"(lofs), "v"(g) : "memory");
}
__device__ __forceinline__ void wait_async_zero() {
  asm volatile("s_wait_asynccnt 0" ::: "memory");
}

// ---------------------------------------------------------------------------
// Fold kernel: W_eff[n][k] = W_org[n][k] + sum_r P'[n][r] * Qc[r][k]
//   P' = [p*lam | -bp*blam]  (128 wide),  Qc = [q ; bq]
// Block: 256 threads, tile = 16 n-rows x 256 k-cols; P'^T broadcast from LDS.
// ---------------------------------------------------------------------------
__global__ __launch_bounds__(256) void tlora_fold_weff(
    const float* __restrict__ Worg,
    const float* __restrict__ qw,  const float* __restrict__ pw,  const float* __restrict__ lam,
    const float* __restrict__ bqw, const float* __restrict__ bpw, const float* __restrict__ blam,
    float* __restrict__ Weff)
{
  __shared__ float Pt[2 * RANK][16];          // [r][n], signed & pre-scaled by lambda
  const int tid = threadIdx.x;
  const int n0  = blockIdx.y * 16;
  const int k   = blockIdx.x * 256 + tid;

  for (int i = tid; i < 2 * RANK * 16; i += 256) {
    int r = i >> 4, n = i & 15;
    float v;
    if (r < RANK) v =  pw[(size_t)(n0 + n) * RANK + r] * lam[r];
    else          v = -bpw[(size_t)(n0 + n) * RANK + (r - RANK)] * blam[r - RANK];
    Pt[r][n] = v;
  }
  __syncthreads();

  float acc[16];
#pragma unroll
  for (int n = 0; n < 16; ++n)
    acc[n] = Worg[(size_t)(n0 + n) * IN_DIM + k];

  for (int r = 0; r < RANK; ++r) {
    float qv = qw[(size_t)r * IN_DIM + k];
#pragma unroll
    for (int n = 0; n < 16; ++n) acc[n] += Pt[r][n] * qv;
  }
  for (int r = 0; r < RANK; ++r) {
    float qv = bqw[(size_t)r * IN_DIM + k];
#pragma unroll
    for (int n = 0; n < 16; ++n) acc[n] += Pt[RANK + r][n] * qv;
  }
#pragma unroll
  for (int n = 0; n < 16; ++n)
    Weff[(size_t)(n0 + n) * IN_DIM + k] = acc[n];
}

// ---------------------------------------------------------------------------
// Main fp32 WMMA GEMM: out[m][n] = sum_k X[m][k] * W[n][k]
// 256 thr (8 waves), block tile 128x128, double-buffered K slabs of 32 in LDS,
// wave tile 64x32 (4x2 WMMA 16x16 tiles).
// A frag: lane l holds X[m0+l%16][kk+2*(l>>4)+{0,1}];  B frag symmetric.
// ---------------------------------------------------------------------------
__global__ __launch_bounds__(256) void tlora_gemm_wmma(
    const float* __restrict__ X, const float* __restrict__ W,
    float* __restrict__ out, int M)
{
  __shared__ float As[2][MT * LDS_STRIDE];
  __shared__ float Bs[2][NT * LDS_STRIDE];

  const int tid  = threadIdx.x;
  const int lane = tid & 31;
  const int wave = tid >> 5;
  const int wm   = wave & 1;        // wave grid 2 (M) x 4 (N)
  const int wn   = wave >> 1;
  const int half = lane >> 4;       // 0 | 1
  const int l16  = lane & 15;

  const int m0 = blockIdx.y * MT;
  const int n0 = blockIdx.x * NT;

  v8f acc[4][2] = {};

  // cooperative global->LDS mapping: thread owns one 16-float half-row
  const int lrow = tid >> 1;              // 0..127
  const int lcol = (tid & 1) * 16;        // 0 | 16
  const float* gA = X + (size_t)(m0 + lrow) * IN_DIM + lcol;
  const float* gB = W + (size_t)(n0 + lrow) * IN_DIM + lcol;
  const int soff = lrow * LDS_STRIDE + lcol;

  // Prologue: start streaming slab 0 into buffer 0.
#pragma unroll
  for (int i = 0; i < 4; ++i) {
    async_g2l_b128(gA + 4 * i, &As[0][soff] + 4 * i);
    async_g2l_b128(gB + 4 * i, &Bs[0][soff] + 4 * i);
  }
  gA += KT;
  gB += KT;

  int cur = 0;
  for (int kt = 0; kt < IN_DIM; kt += KT) {
    wait_async_zero();       // my slab-'cur' transfers are in LDS
    __syncthreads();         // everyone's are; everyone also done reading buf cur^1

    if (kt + KT < IN_DIM) {  // stream next slab into the other buffer during compute
      float* sA = &As[cur ^ 1][soff];
      float* sB = &Bs[cur ^ 1][soff];
#pragma unroll
      for (int i = 0; i < 4; ++i) {
        async_g2l_b128(gA + 4 * i, sA + 4 * i);
        async_g2l_b128(gB + 4 * i, sB + 4 * i);
      }
      __builtin_prefetch(gA + KT, 0, 1);   // warm L2 two slabs ahead
      __builtin_prefetch(gB + KT, 0, 1);
      gA += KT;
      gB += KT;
    }

    const float* a_base = &As[cur][(wm * 64 + l16) * LDS_STRIDE];
    const float* b_base = &Bs[cur][(wn * 32 + l16) * LDS_STRIDE];

#pragma unroll
    for (int kk = 0; kk < KT; kk += 4) {
      const int col = kk + 2 * half;
      v2f af[4], bf[2];
#pragma unroll
      for (int im = 0; im < 4; ++im)
        af[im] = *(const v2f*)(a_base + im * 16 * LDS_STRIDE + col);
#pragma unroll
      for (int in = 0; in < 2; ++in)
        bf[in] = *(const v2f*)(b_base + in * 16 * LDS_STRIDE + col);
#pragma unroll
      for (int im = 0; im < 4; ++im)
#pragma unroll
        for (int in = 0; in < 2; ++in)
          acc[im][in] = __builtin_amdgcn_wmma_f32_16x16x4_f32(
              /*neg_a=*/false, af[im], /*neg_b=*/false, bf[in],
              /*c_mod=*/(short)0, acc[im][in],
              /*reuse_a=*/false, /*reuse_b=*/false);
    }
    cur ^= 1;
  }

  // Epilogue: C/D layout -> VGPR j holds M = tile_m + half*8 + j, N = tile_n + l16
#pragma unroll
  for (int im = 0; im < 4; ++im) {
#pragma unroll
    for (int in = 0; in < 2; ++in) {
      const int mbase = m0 + wm * 64 + im * 16 + half * 8;
      const int nidx  = n0 + wn * 32 + in * 16 + l16;
#pragma unroll
      for (int j = 0; j < 8; ++j)
        out[(size_t)(mbase + j) * OUT_DIM + nidx] = acc[im][in][j];
    }
  }
}

// ---------------------------------------------------------------------------
extern "C" void kernel_launch(void* const* d_in, const int* in_sizes, int n_in,
                              void* d_out, int out_size, void* d_ws, size_t ws_size,
                              hipStream_t stream) {
  const float* x     = (const float*)d_in[0];
  const float* W_org = (const float*)d_in[1];
  const float* q_w   = (const float*)d_in[2];
  const float* p_w   = (const float*)d_in[3];
  const float* lam   = (const float*)d_in[4];
  const float* bq_w  = (const float*)d_in[5];
  const float* bp_w  = (const float*)d_in[6];
  const float* blam  = (const float*)d_in[7];
  float* out = (float*)d_out;

  const int M = in_sizes[0] / IN_DIM;   // 16384

  const size_t weff_bytes = (size_t)OUT_DIM * IN_DIM * sizeof(float);  // 64MB
  const float* Wuse = W_org;
  if (ws_size >= weff_bytes) {
    float* Weff = (float*)d_ws;
    tlora_fold_weff<<<dim3(IN_DIM / 256, OUT_DIM / 16), 256, 0, stream>>>(
        W_org, q_w, p_w, lam, bq_w, bp_w, blam, Weff);
    Wuse = Weff;
  }

  tlora_gemm_wmma<<<dim3(OUT_DIM / NT, M / MT), 256, 0, stream>>>(x, Wuse, out, M);
}